// GCN_14869176778981
// MI455X (gfx1250) — compile-verified
//
#include <hip/hip_runtime.h>

typedef float v2f __attribute__((ext_vector_type(2)));
typedef float v4f __attribute__((ext_vector_type(4)));
typedef float v8f __attribute__((ext_vector_type(8)));

// ---------------- degree / dinv ----------------

__global__ void fill_ones_k(float* __restrict__ p, int n) {
    int i = blockIdx.x * blockDim.x + threadIdx.x;
    if (i < n) p[i] = 1.0f;
}

__global__ void deg_accum_k(float* __restrict__ deg, const int* __restrict__ dst, int e) {
    int i = blockIdx.x * blockDim.x + threadIdx.x;
    if (i < e) unsafeAtomicAdd(&deg[dst[i]], 1.0f);
}

__global__ void to_rsqrt_k(float* __restrict__ p, int n) {
    int i = blockIdx.x * blockDim.x + threadIdx.x;
    if (i < n) p[i] = __frsqrt_rn(p[i]);
}

// ---------------- WMMA GEMM:  H[n,OUT] = (relu?)A[n,K] @ W[K,OUT] ----------------
// One wave computes one 16x16 output tile (OUT=64 -> 4 waves/block; OUT=7 -> 1 wave).
// A tile staged into LDS via CDNA5 async global->LDS b128 copies (ASYNCcnt),
// pitch K+4 floats (bank-conflict-avoiding, keeps 16B alignment).
// ReLU of the previous layer's output is applied at LDS fragment-read time.

template <int K, int OUT, bool RELU>
__global__ __launch_bounds__(128) void gemm_wmma_k(const float* __restrict__ A,
                                                   const float* __restrict__ W,
                                                   float* __restrict__ H, int n) {
    constexpr int PITCH = K + 4;
    __shared__ float As[16 * PITCH];

    const int row0 = blockIdx.x * 16;

    // Async-stage the 16xK A tile: each lane moves one 16B chunk per iteration.
    // GVS addressing: 64-bit SGPR base (A) + 32-bit VGPR byte offset (< 2^31 here).
    for (int idx = threadIdx.x; idx < 16 * (K / 4); idx += blockDim.x) {
        int r  = idx / (K / 4);
        int c4 = (idx - r * (K / 4)) * 4;
        int row = row0 + r;
        if (row >= n) row = n - 1;  // n is a multiple of 16 in practice; safe clamp
        unsigned lds_off = (unsigned)((r * PITCH + c4) * 4);
        unsigned mem_off = (unsigned)(((size_t)row * K + c4) * 4);
        unsigned lds_addr = (unsigned)(size_t)&As[0] + lds_off;  // LDS byte address
        asm volatile("global_load_async_to_lds_b128 %0, %1, %2"
                     :
                     : "v"(lds_addr), "v"(mem_off), "s"(A)
                     : "memory");
    }
    asm volatile("s_wait_asynccnt 0x0" ::: "memory");
    __syncthreads();

    const int wave = threadIdx.x >> 5;
    const int lane = threadIdx.x & 31;
    const int half = lane >> 4;   // 0: K pair {0,1}; 1: K pair {2,3}
    const int l16  = lane & 15;
    const int c0   = wave * 16;
    const int col  = c0 + l16;
    const int colc = (col < OUT) ? col : (OUT - 1);  // clamp W reads for OUT=7 padding

    v8f acc = {};
#pragma unroll
    for (int kk = 0; kk < K; kk += 4) {
        const int kb = kk + 2 * half;
        v2f a;
        a.x = As[l16 * PITCH + kb];
        a.y = As[l16 * PITCH + kb + 1];
        if (RELU) {
            a.x = fmaxf(a.x, 0.0f);
            a.y = fmaxf(a.y, 0.0f);
        }
        v2f b;
        b.x = W[(size_t)kb * OUT + colc];
        b.y = W[(size_t)(kb + 1) * OUT + colc];
        acc = __builtin_amdgcn_wmma_f32_16x16x4_f32(false, a, false, b,
                                                    (short)0, acc, false, false);
    }

    if (col < OUT) {
#pragma unroll
        for (int r = 0; r < 8; r++) {
            int m = row0 + r + 8 * half;
            if (m < n) H[(size_t)m * OUT + col] = acc[r];
        }
    }
}

// ---------------- agg init: agg = h * dinv^2 + b ----------------

template <int F>
__global__ void init_agg_k(float* __restrict__ agg, const float* __restrict__ h,
                           const float* __restrict__ dinv, const float* __restrict__ bias,
                           int n) {
    int i = blockIdx.x * blockDim.x + threadIdx.x;
    if (i >= n * F) return;
    int node = i / F;
    int f    = i - node * F;
    float di = dinv[node];
    agg[i] = fmaf(h[i], di * di, bias[f]);
}

// ---------------- edge scatter (F=64): one wave per edge, float2 per lane ----------------

__global__ void scatter64_k(const float* __restrict__ h, const int* __restrict__ src,
                            const int* __restrict__ dst, const float* __restrict__ dinv,
                            float* __restrict__ agg, int e) {
    int t    = blockIdx.x * blockDim.x + threadIdx.x;
    int ed   = t >> 5;
    int lane = t & 31;
    if (ed >= e) return;
    int s = src[ed];
    int d = dst[ed];
    float coef = dinv[s] * dinv[d];
    const float2* hp = (const float2*)(h + (size_t)s * 64);
    float2 v = hp[lane];
    float* out = agg + (size_t)d * 64 + lane * 2;
    unsafeAtomicAdd(out + 0, v.x * coef);
    unsafeAtomicAdd(out + 1, v.y * coef);
}

// ---------------- edge scatter (F=7): 8 lanes per edge, lane 7 idle ----------------

__global__ void scatter7_k(const float* __restrict__ h, const int* __restrict__ src,
                           const int* __restrict__ dst, const float* __restrict__ dinv,
                           float* __restrict__ out, int e) {
    int t  = blockIdx.x * blockDim.x + threadIdx.x;
    int ed = t >> 3;
    int f  = t & 7;
    if (ed >= e || f >= 7) return;
    int s = src[ed];
    int d = dst[ed];
    float coef = dinv[s] * dinv[d];
    unsafeAtomicAdd(&out[(size_t)d * 7 + f], h[(size_t)s * 7 + f] * coef);
}

// ---------------- driver ----------------

extern "C" void kernel_launch(void* const* d_in, const int* in_sizes, int n_in,
                              void* d_out, int out_size, void* d_ws, size_t ws_size,
                              hipStream_t stream) {
    const float* x  = (const float*)d_in[0];
    const int*   ei = (const int*)d_in[1];
    const float* W1 = (const float*)d_in[2];
    const float* b1 = (const float*)d_in[3];
    const float* W2 = (const float*)d_in[4];
    const float* b2 = (const float*)d_in[5];
    const float* W3 = (const float*)d_in[6];
    const float* b3 = (const float*)d_in[7];

    const int n = in_sizes[0] / 128;
    const int e = in_sizes[1] / 2;
    const int* src = ei;
    const int* dst = ei + e;
    float* out = (float*)d_out;

    // workspace layout
    float* dinv = (float*)d_ws;
    size_t nAl  = ((size_t)n + 63) & ~(size_t)63;
    float* hbuf = dinv + nAl;               // n*64
    float* abuf = hbuf + (size_t)n * 64;    // n*64
    float* h3   = abuf + (size_t)n * 64;    // n*7

    const int TB = 256;
    const int rowTiles = (n + 15) / 16;

    // symmetric-normalization coefficients (shared by all layers)
    fill_ones_k<<<(n + TB - 1) / TB, TB, 0, stream>>>(dinv, n);
    deg_accum_k<<<(e + TB - 1) / TB, TB, 0, stream>>>(dinv, dst, e);
    to_rsqrt_k<<<(n + TB - 1) / TB, TB, 0, stream>>>(dinv, n);

    // Layer 1: 128 -> 64
    gemm_wmma_k<128, 64, false><<<rowTiles, 128, 0, stream>>>(x, W1, hbuf, n);
    init_agg_k<64><<<(n * 64 + TB - 1) / TB, TB, 0, stream>>>(abuf, hbuf, dinv, b1, n);
    scatter64_k<<<(int)(((size_t)e * 32 + TB - 1) / TB), TB, 0, stream>>>(hbuf, src, dst, dinv, abuf, e);

    // Layer 2: 64 -> 64 (ReLU of layer-1 output folded into GEMM A fragment read)
    gemm_wmma_k<64, 64, true><<<rowTiles, 128, 0, stream>>>(abuf, W2, hbuf, n);
    init_agg_k<64><<<(n * 64 + TB - 1) / TB, TB, 0, stream>>>(abuf, hbuf, dinv, b2, n);
    scatter64_k<<<(int)(((size_t)e * 32 + TB - 1) / TB), TB, 0, stream>>>(hbuf, src, dst, dinv, abuf, e);

    // Layer 3: 64 -> 7 (single wave per row tile, masked column stores)
    gemm_wmma_k<64, 7, true><<<rowTiles, 32, 0, stream>>>(abuf, W3, h3, n);
    init_agg_k<7><<<(n * 7 + TB - 1) / TB, TB, 0, stream>>>(out, h3, dinv, b3, n);
    scatter7_k<<<(int)(((size_t)e * 8 + TB - 1) / TB), TB, 0, stream>>>(h3, src, dst, dinv, out, e);
}